// HBGATPN_45569603010826
// MI455X (gfx1250) — compile-verified
//
#include <hip/hip_runtime.h>
#include <cstddef>

#define HIDD 128
#define HHEADS 4
#define HC 512          // HHEADS * HIDD
#define NTASK 20000
#define NSTATION 1000
#define NWORKER 2000
#define NGRAPHS 64
#define EPSF 1e-16f

typedef __attribute__((ext_vector_type(2))) float v2f;
typedef __attribute__((ext_vector_type(8))) float v8f;

// ---------------- order-preserving float<->uint map for atomicMax ----------
__device__ __forceinline__ unsigned mapf(float f) {
  unsigned u = __float_as_uint(f);
  return (u & 0x80000000u) ? ~u : (u | 0x80000000u);
}
__device__ __forceinline__ float unmapf(unsigned u) {
  return __uint_as_float((u & 0x80000000u) ? (u ^ 0x80000000u) : ~u);
}

// ---------------- embed: relu(layernorm(x@W + b) * g + beta) ---------------
// block = 128 threads (one row per block)
__global__ void k_embed(const float* __restrict__ x, const float* __restrict__ W,
                        const float* __restrict__ b, const float* __restrict__ g,
                        const float* __restrict__ beta, float* __restrict__ y,
                        int N, int Fin) {
  int row = blockIdx.x;
  int c = threadIdx.x;
  __shared__ float xs[32];
  __shared__ float red[4];
  if (c < Fin) xs[c] = x[(size_t)row * Fin + c];
  __syncthreads();
  float acc = b[c];
  for (int k = 0; k < Fin; ++k) acc += xs[k] * W[k * HIDD + c];
  // mean
  float v = acc;
  for (int o = 16; o > 0; o >>= 1) v += __shfl_xor(v, o, 32);
  int wid = c >> 5, lane = c & 31;
  if (lane == 0) red[wid] = v;
  __syncthreads();
  float mean = (red[0] + red[1] + red[2] + red[3]) * (1.0f / HIDD);
  __syncthreads();
  float d = acc - mean, d2 = d * d;
  for (int o = 16; o > 0; o >>= 1) d2 += __shfl_xor(d2, o, 32);
  if (lane == 0) red[wid] = d2;
  __syncthreads();
  float var = (red[0] + red[1] + red[2] + red[3]) * (1.0f / HIDD);
  float ln = d * rsqrtf(var + 1e-5f) * g[c] + beta[c];
  y[(size_t)row * HIDD + c] = fmaxf(ln, 0.0f);
}

// ---------------- WMMA fp32 GEMM: Y[N,512] = X[N,128] @ W[128,512] + bias --
// block = (32,4): 4 waves, each computes a 32-row x 64-col tile.
// A rows are index-clamped (not predicated) so the hot loop has no exec-mask
// manipulation; out-of-range rows compute garbage that is never stored.
__global__ void k_gemm(const float* __restrict__ X, const float* __restrict__ W,
                       const float* __restrict__ bias, float* __restrict__ Y, int N) {
  int lane = threadIdx.x;
  int rt = blockIdx.y * blockDim.y + threadIdx.y;
  int rbase = rt * 32;
  if (rbase >= N) return;
  int c0 = blockIdx.x * 64;
  int koff = (lane >= 16) ? 2 : 0;
  int m15 = lane & 15;
  int row0 = rbase + m15;       if (row0 > N - 1) row0 = N - 1;
  int row1 = rbase + 16 + m15;  if (row1 > N - 1) row1 = N - 1;
  const float* arow0 = X + (size_t)row0 * HIDD + koff;
  const float* arow1 = X + (size_t)row1 * HIDD + koff;
  v8f a00 = {0,0,0,0,0,0,0,0};
  v8f a01 = a00, a02 = a00, a03 = a00;   // rows rbase..rbase+15
  v8f a10 = a00, a11 = a00, a12 = a00, a13 = a00;  // rows rbase+16..rbase+31
#pragma unroll 4
  for (int k = 0; k < HIDD; k += 4) {
    v2f av0; av0.x = arow0[k]; av0.y = arow0[k + 1];
    v2f av1; av1.x = arow1[k]; av1.y = arow1[k + 1];
    const float* wp = W + (size_t)(k + koff) * HC + c0 + m15;
    v2f b0; b0.x = wp[0];  b0.y = wp[HC];
    v2f b1; b1.x = wp[16]; b1.y = wp[16 + HC];
    v2f b2; b2.x = wp[32]; b2.y = wp[32 + HC];
    v2f b3; b3.x = wp[48]; b3.y = wp[48 + HC];
    a00 = __builtin_amdgcn_wmma_f32_16x16x4_f32(false, av0, false, b0, (short)0, a00, false, false);
    a10 = __builtin_amdgcn_wmma_f32_16x16x4_f32(false, av1, false, b0, (short)0, a10, false, false);
    a01 = __builtin_amdgcn_wmma_f32_16x16x4_f32(false, av0, false, b1, (short)0, a01, false, false);
    a11 = __builtin_amdgcn_wmma_f32_16x16x4_f32(false, av1, false, b1, (short)0, a11, false, false);
    a02 = __builtin_amdgcn_wmma_f32_16x16x4_f32(false, av0, false, b2, (short)0, a02, false, false);
    a12 = __builtin_amdgcn_wmma_f32_16x16x4_f32(false, av1, false, b2, (short)0, a12, false, false);
    a03 = __builtin_amdgcn_wmma_f32_16x16x4_f32(false, av0, false, b3, (short)0, a03, false, false);
    a13 = __builtin_amdgcn_wmma_f32_16x16x4_f32(false, av1, false, b3, (short)0, a13, false, false);
  }
  int rlo = rbase + ((lane >= 16) ? 8 : 0);
  float bb0 = bias[c0 + m15], bb1 = bias[c0 + 16 + m15];
  float bb2 = bias[c0 + 32 + m15], bb3 = bias[c0 + 48 + m15];
#pragma unroll
  for (int j = 0; j < 8; ++j) {
    int r = rlo + j;
    if (r < N) {
      float* yp = Y + (size_t)r * HC + c0 + m15;
      yp[0]  = a00[j] + bb0;
      yp[16] = a01[j] + bb1;
      yp[32] = a02[j] + bb2;
      yp[48] = a03[j] + bb3;
    }
    int r2 = r + 16;
    if (r2 < N) {
      float* yp = Y + (size_t)r2 * HC + c0 + m15;
      yp[0]  = a10[j] + bb0;
      yp[16] = a11[j] + bb1;
      yp[32] = a12[j] + bb2;
      yp[48] = a13[j] + bb3;
    }
  }
}

// ---------------- utility fills ----------------
__global__ void k_fill(float* __restrict__ p, float v, size_t n) {
  size_t i = (size_t)blockIdx.x * blockDim.x + threadIdx.x;
  size_t s = (size_t)gridDim.x * blockDim.x;
  for (; i < n; i += s) p[i] = v;
}

// out[i*128+c] = b1[c] (+ b2[c]) (+ b3[c])
__global__ void k_init_bias(float* __restrict__ out, const float* __restrict__ b1,
                            const float* __restrict__ b2, const float* __restrict__ b3,
                            int N) {
  size_t i = (size_t)blockIdx.x * blockDim.x + threadIdx.x;
  size_t tot = (size_t)N * HIDD;
  size_t s = (size_t)gridDim.x * blockDim.x;
  for (; i < tot; i += s) {
    int c = (int)(i & (HIDD - 1));
    float v = b1[c];
    if (b2) v += b2[c];
    if (b3) v += b3[c];
    out[i] = v;
  }
}

// ---------------- GATv2 edge pass 1: logits + segment max -----------------
// one wave per edge (grid-stride), 4 heads per wave
__global__ void k_edge_logits(const float* __restrict__ XL, const float* __restrict__ XR,
                              const int* __restrict__ src, const int* __restrict__ dst,
                              const float* __restrict__ att, float* __restrict__ escr,
                              unsigned* __restrict__ mbuf, int E) {
  int lane = threadIdx.x & 31;
  int gw = (int)((blockIdx.x * blockDim.x + threadIdx.x) >> 5);
  int nw = (int)((gridDim.x * blockDim.x) >> 5);
  for (int e = gw; e < E; e += nw) {
    int s = src[e], d = dst[e];
    const float4* pl = (const float4*)(XL + (size_t)s * HC);
    const float4* pr = (const float4*)(XR + (size_t)d * HC);
    const float4* pa = (const float4*)att;
#pragma unroll
    for (int h = 0; h < HHEADS; ++h) {
      float4 l = pl[h * 32 + lane];
      float4 r = pr[h * 32 + lane];
      float4 a = pa[h * 32 + lane];
      float t0 = l.x + r.x, t1 = l.y + r.y, t2 = l.z + r.z, t3 = l.w + r.w;
      t0 = t0 > 0.f ? t0 : 0.2f * t0;
      t1 = t1 > 0.f ? t1 : 0.2f * t1;
      t2 = t2 > 0.f ? t2 : 0.2f * t2;
      t3 = t3 > 0.f ? t3 : 0.2f * t3;
      float sx = a.x * t0 + a.y * t1 + a.z * t2 + a.w * t3;
      for (int o = 16; o > 0; o >>= 1) sx += __shfl_xor(sx, o, 32);
      if (lane == 0) {
        escr[(size_t)e * HHEADS + h] = sx;
        atomicMax(&mbuf[(size_t)d * HHEADS + h], mapf(sx));
      }
    }
  }
}

// ---------------- pass 2: ex = exp(logit - m); den += ex ------------------
__global__ void k_edge_exp(const int* __restrict__ dst, const unsigned* __restrict__ mbuf,
                           float* __restrict__ escr, float* __restrict__ den, int E) {
  size_t i = (size_t)blockIdx.x * blockDim.x + threadIdx.x;
  size_t tot = (size_t)E * HHEADS;
  size_t s = (size_t)gridDim.x * blockDim.x;
  for (; i < tot; i += s) {
    int e = (int)(i >> 2), h = (int)(i & 3);
    int d = dst[e];
    float m = unmapf(mbuf[(size_t)d * HHEADS + h]);
    float ex = __expf(escr[i] - m);
    escr[i] = ex;
    atomicAdd(&den[(size_t)d * HHEADS + h], ex);
  }
}

// ---------------- pass 3: out[dst] += mean_h(alpha_h * xl[src,h,:]) -------
// block = 128 threads, one edge per block
__global__ void k_edge_scatter(const float* __restrict__ XL, const int* __restrict__ src,
                               const int* __restrict__ dst, const float* __restrict__ escr,
                               const float* __restrict__ den, float* __restrict__ out, int E) {
  int e = blockIdx.x;
  if (e >= E) return;
  int c = threadIdx.x;
  int s = src[e], d = dst[e];
  float acc = 0.0f;
#pragma unroll
  for (int h = 0; h < HHEADS; ++h) {
    float alpha = escr[(size_t)e * HHEADS + h] / (den[(size_t)d * HHEADS + h] + EPSF);
    acc += alpha * XL[(size_t)s * HC + h * HIDD + c];
  }
  atomicAdd(&out[(size_t)d * HIDD + c], 0.25f * acc);
}

// ---------------- residual: x += relu(nx) ----------------
__global__ void k_relu_res(float* __restrict__ x, const float* __restrict__ nx, size_t n) {
  size_t i = (size_t)blockIdx.x * blockDim.x + threadIdx.x;
  size_t s = (size_t)gridDim.x * blockDim.x;
  for (; i < n; i += s) x[i] += fmaxf(nx[i], 0.0f);
}

// ---------------- pool pass 1: score = mlp2(x); segment max ---------------
// one wave per node (grid-stride); lane j computes hidden unit j (32 hidden)
__global__ void k_score(const float* __restrict__ x, const float* __restrict__ W1,
                        const float* __restrict__ b1, const float* __restrict__ W2,
                        const float* __restrict__ b2, const int* __restrict__ batch,
                        float* __restrict__ scores, unsigned* __restrict__ gm, int N) {
  int lane = threadIdx.x & 31;
  int gw = (int)((blockIdx.x * blockDim.x + threadIdx.x) >> 5);
  int nw = (int)((gridDim.x * blockDim.x) >> 5);
  for (int i = gw; i < N; i += nw) {
    const float* xr = x + (size_t)i * HIDD;
    float h = b1[lane];
    for (int k = 0; k < HIDD; ++k) h += xr[k] * W1[k * 32 + lane];
    h = fmaxf(h, 0.0f);
    float p = h * W2[lane];
    for (int o = 16; o > 0; o >>= 1) p += __shfl_xor(p, o, 32);
    float score = p + b2[0];
    if (lane == 0) {
      scores[i] = score;
      atomicMax(&gm[batch[i]], mapf(score));
    }
  }
}

__global__ void k_pool_exp(const int* __restrict__ batch, const unsigned* __restrict__ gm,
                           float* __restrict__ scores, float* __restrict__ gden, int N) {
  int i = blockIdx.x * blockDim.x + threadIdx.x;
  int s = gridDim.x * blockDim.x;
  for (; i < N; i += s) {
    int g = batch[i];
    float ex = __expf(scores[i] - unmapf(gm[g]));
    scores[i] = ex;
    atomicAdd(&gden[g], ex);
  }
}

// block = 128 threads, one node per block; accumulates into gctx columns
__global__ void k_pool_scatter(const float* __restrict__ x, const int* __restrict__ batch,
                               const float* __restrict__ scores, const float* __restrict__ gden,
                               float* __restrict__ gctx, int colbase, int N) {
  int i = blockIdx.x;
  if (i >= N) return;
  int c = threadIdx.x;
  int g = batch[i];
  float alpha = scores[i] / (gden[g] + EPSF);
  atomicAdd(&gctx[(size_t)g * 384 + colbase + c], x[(size_t)i * HIDD + c] * alpha);
}

// ---------------- critic: value = relu(gctx@W1+b1)@W2 + b2 ----------------
// block = 64 threads, one graph per block
__global__ void k_critic(const float* __restrict__ gctx, const float* __restrict__ W1,
                         const float* __restrict__ b1, const float* __restrict__ W2,
                         const float* __restrict__ b2, float* __restrict__ value) {
  int g = blockIdx.x;
  int j = threadIdx.x;
  __shared__ float sh[64];
  const float* xr = gctx + (size_t)g * 384;
  float h = b1[j];
  for (int k = 0; k < 384; ++k) h += xr[k] * W1[k * 64 + j];
  h = fmaxf(h, 0.0f);
  sh[j] = h * W2[j];
  __syncthreads();
  for (int o = 32; o > 0; o >>= 1) {
    if (j < o) sh[j] += sh[j + o];
    __syncthreads();
  }
  if (j == 0) value[g] = sh[0] + b2[0];
}

// ======================= host orchestration =======================
static inline int cdiv(int a, int b) { return (a + b - 1) / b; }

static void run_gemm(hipStream_t st, const float* X, const float* W, const float* b,
                     float* Y, int N) {
  dim3 grid(8, (unsigned)cdiv(cdiv(N, 32), 4));
  dim3 blk(32, 4);
  k_gemm<<<grid, blk, 0, st>>>(X, W, b, Y, N);
}

static void run_gat(hipStream_t st, const float* xsrc, const float* xdst, int nsrc, int ndst,
                    const int* src, const int* dst, int E,
                    const float* Wl, const float* bl, const float* Wr, const float* br,
                    const float* att, float* out,
                    float* XL, float* XR, float* escr, unsigned* mbuf, float* den) {
  run_gemm(st, xsrc, Wl, bl, XL, nsrc);
  run_gemm(st, xdst, Wr, br, XR, ndst);
  size_t md = (size_t)ndst * HHEADS;
  k_fill<<<cdiv((int)md, 256), 256, 0, st>>>((float*)mbuf, 0.0f, md);  // 0u == mapf lower bound
  k_fill<<<cdiv((int)md, 256), 256, 0, st>>>(den, 0.0f, md);
  k_edge_logits<<<cdiv(E, 8), 256, 0, st>>>(XL, XR, src, dst, att, escr, mbuf, E);
  k_edge_exp<<<cdiv(E * 4, 256), 256, 0, st>>>(dst, mbuf, escr, den, E);
  k_edge_scatter<<<E, 128, 0, st>>>(XL, src, dst, escr, den, out, E);
}

static void run_pool(hipStream_t st, const float* x, const int* batch, int N,
                     const float* W1, const float* b1, const float* W2, const float* b2,
                     float* gctx, int colbase, float* scores, unsigned* gm, float* gden) {
  k_fill<<<1, 64, 0, st>>>((float*)gm, 0.0f, NGRAPHS);
  k_fill<<<1, 64, 0, st>>>(gden, 0.0f, NGRAPHS);
  k_score<<<cdiv(N, 8), 256, 0, st>>>(x, W1, b1, W2, b2, batch, scores, gm, N);
  k_pool_exp<<<cdiv(N, 256), 256, 0, st>>>(batch, gm, scores, gden, N);
  k_pool_scatter<<<N, 128, 0, st>>>(x, batch, scores, gden, gctx, colbase, N);
}

extern "C" void kernel_launch(void* const* d_in, const int* in_sizes, int n_in,
                              void* d_out, int out_size, void* d_ws, size_t ws_size,
                              hipStream_t stream) {
  (void)in_sizes; (void)n_in; (void)out_size; (void)ws_size;
  const float* x_task    = (const float*)d_in[0];
  const float* x_worker  = (const float*)d_in[1];
  const float* x_station = (const float*)d_in[2];
  // edges: tt, at, ht, cd, db  (src,dst pairs)
  const int* tt_s = (const int*)d_in[3];  const int* tt_d = (const int*)d_in[4];
  const int* at_s = (const int*)d_in[5];  const int* at_d = (const int*)d_in[6];
  const int* ht_s = (const int*)d_in[7];  const int* ht_d = (const int*)d_in[8];
  const int* cd_s = (const int*)d_in[9];  const int* cd_d = (const int*)d_in[10];
  const int* db_s = (const int*)d_in[11]; const int* db_d = (const int*)d_in[12];
  const int* bat_t = (const int*)d_in[13];
  const int* bat_s = (const int*)d_in[14];
  const int* bat_w = (const int*)d_in[15];
  // embeddings: task(16..19), worker(20..23), station(24..27): {W,b,g,beta}
  // layers: base 28 + l*30 + et*6, et order tt,at,ht,cd,db; {Wl,bl,Wr,br,att,bias}
  // station_attn 88..91, tw_attn 92..95, critic 96..99 : {W1,b1,W2,b2}

  // ---- d_out layout: t | s | w | gctx | value ----
  float* t = (float*)d_out;                                   // [20000,128]
  float* s = t + (size_t)NTASK * HIDD;                        // [1000,128]
  float* w = s + (size_t)NSTATION * HIDD;                     // [2000,128]
  float* gctx = w + (size_t)NWORKER * HIDD;                   // [64,384]
  float* value = gctx + (size_t)NGRAPHS * 384;                // [64,1]

  // ---- workspace carve-up (floats) ----
  float* ws = (float*)d_ws;
  float* XL   = ws;                              // 20000*512
  float* XR   = XL + (size_t)NTASK * HC;         // 20000*512
  float* nt   = XR + (size_t)NTASK * HC;         // 20000*128
  float* ns   = nt + (size_t)NTASK * HIDD;       // 1000*128
  float* nw   = ns + (size_t)NSTATION * HIDD;    // 2000*128
  float* escr = nw + (size_t)NWORKER * HIDD;     // 320000*4
  float* mbuf_f = escr + (size_t)320000 * HHEADS; // 20000*4
  float* den  = mbuf_f + (size_t)NTASK * HHEADS; // 20000*4
  float* scores = den + (size_t)NTASK * HHEADS;  // 20000
  float* gm_f = scores + NTASK;                  // 64
  float* gden = gm_f + NGRAPHS;                  // 64
  unsigned* mbuf = (unsigned*)mbuf_f;
  unsigned* gm = (unsigned*)gm_f;

  // ---- embeddings (write live node features directly into d_out) ----
  {
    const float* p = (const float*)d_in[16];
    k_embed<<<NTASK, 128, 0, stream>>>(x_task, p, (const float*)d_in[17],
                                       (const float*)d_in[18], (const float*)d_in[19],
                                       t, NTASK, 32);
    k_embed<<<NWORKER, 128, 0, stream>>>(x_worker, (const float*)d_in[20],
                                         (const float*)d_in[21], (const float*)d_in[22],
                                         (const float*)d_in[23], w, NWORKER, 16);
    k_embed<<<NSTATION, 128, 0, stream>>>(x_station, (const float*)d_in[24],
                                          (const float*)d_in[25], (const float*)d_in[26],
                                          (const float*)d_in[27], s, NSTATION, 16);
  }

  // ---- 2 hetero GATv2 layers ----
  for (int l = 0; l < 2; ++l) {
    int base = 28 + l * 30;
    const float** P = (const float**)(d_in + base);  // 30 consecutive params
    // et offsets: tt=0, at=6, ht=12, cd=18, db=24
    const float* tt_bias = P[0 + 5];
    const float* at_bias = P[6 + 5];
    const float* ht_bias = P[12 + 5];
    const float* cd_bias = P[18 + 5];
    const float* db_bias = P[24 + 5];
    k_init_bias<<<2048, 256, 0, stream>>>(nt, tt_bias, ht_bias, cd_bias, NTASK);
    k_init_bias<<<512, 256, 0, stream>>>(ns, at_bias, nullptr, nullptr, NSTATION);
    k_init_bias<<<1024, 256, 0, stream>>>(nw, db_bias, nullptr, nullptr, NWORKER);

    // tt: t -> t
    run_gat(stream, t, t, NTASK, NTASK, tt_s, tt_d, 320000,
            P[0], P[1], P[2], P[3], P[4], nt, XL, XR, escr, mbuf, den);
    // ht: s -> t
    run_gat(stream, s, t, NSTATION, NTASK, ht_s, ht_d, 20000,
            P[12], P[13], P[14], P[15], P[16], nt, XL, XR, escr, mbuf, den);
    // cd: w -> t
    run_gat(stream, w, t, NWORKER, NTASK, cd_s, cd_d, 160000,
            P[18], P[19], P[20], P[21], P[22], nt, XL, XR, escr, mbuf, den);
    // at: t -> s
    run_gat(stream, t, s, NTASK, NSTATION, at_s, at_d, 20000,
            P[6], P[7], P[8], P[9], P[10], ns, XL, XR, escr, mbuf, den);
    // db: t -> w
    run_gat(stream, t, w, NTASK, NWORKER, db_s, db_d, 160000,
            P[24], P[25], P[26], P[27], P[28], nw, XL, XR, escr, mbuf, den);

    // residual + relu (after ALL edge types of this layer consumed inputs)
    k_relu_res<<<2048, 256, 0, stream>>>(t, nt, (size_t)NTASK * HIDD);
    k_relu_res<<<512, 256, 0, stream>>>(s, ns, (size_t)NSTATION * HIDD);
    k_relu_res<<<1024, 256, 0, stream>>>(w, nw, (size_t)NWORKER * HIDD);
  }

  // ---- pooling into gctx (zero gctx+value region first: d_out is poisoned) --
  k_fill<<<cdiv(NGRAPHS * 384 + NGRAPHS, 256), 256, 0, stream>>>(
      gctx, 0.0f, (size_t)NGRAPHS * 384 + NGRAPHS);
  const float* sa_W1 = (const float*)d_in[88]; const float* sa_b1 = (const float*)d_in[89];
  const float* sa_W2 = (const float*)d_in[90]; const float* sa_b2 = (const float*)d_in[91];
  const float* tw_W1 = (const float*)d_in[92]; const float* tw_b1 = (const float*)d_in[93];
  const float* tw_W2 = (const float*)d_in[94]; const float* tw_b2 = (const float*)d_in[95];
  run_pool(stream, s, bat_s, NSTATION, sa_W1, sa_b1, sa_W2, sa_b2, gctx, 0,   scores, gm, gden);
  run_pool(stream, t, bat_t, NTASK,    tw_W1, tw_b1, tw_W2, tw_b2, gctx, 128, scores, gm, gden);
  run_pool(stream, w, bat_w, NWORKER,  tw_W1, tw_b1, tw_W2, tw_b2, gctx, 256, scores, gm, gden);

  // ---- critic ----
  k_critic<<<NGRAPHS, 64, 0, stream>>>(gctx, (const float*)d_in[96], (const float*)d_in[97],
                                       (const float*)d_in[98], (const float*)d_in[99], value);
}